// EfficientAttention_6090263626326
// MI455X (gfx1250) — compile-verified
//
#include <hip/hip_runtime.h>
#include <hip/hip_bf16.h>
#include <math.h>

typedef float v2f __attribute__((ext_vector_type(2)));
typedef float v8f __attribute__((ext_vector_type(8)));

#define WMMA_F32_4(A, B, C) \
  __builtin_amdgcn_wmma_f32_16x16x4_f32(false, (A), false, (B), (short)0, (C), false, false)

constexpr int   kB     = 2;
constexpr int   kN     = 16384;   // 128*128 tokens
constexpr int   kHW    = 128;
constexpr int   kHeads = 8;
constexpr int   kHD    = 8;
constexpr int   kNK    = 256;     // reduced kv tokens (16*16)
constexpr float kScale = 0.35355339059327378f;  // 8^-0.5

// ---------------------------------------------------------------------------
// Generic [rows,64] @ [64,64] (+bias) GEMM with V_WMMA_F32_16X16X4_F32.
// Block = 128 threads = 4 waves; wave w computes output cols [16w,16w+16)
// for the block's 16 rows. 16 chained K=4 WMMAs cover K=64.
// ---------------------------------------------------------------------------
__global__ __launch_bounds__(128) void gemm64_wmma(
    const float* __restrict__ X, const float* __restrict__ W,
    const float* __restrict__ bias, float* __restrict__ Y) {
  const int tid  = threadIdx.x;
  const int wv   = tid >> 5;
  const int lane = tid & 31;
  const int lo   = lane & 15;
  const int hi   = lane >> 4;
  const int rowBase = blockIdx.x * 16;
  const int cc      = wv * 16 + lo;

  const float* xr = X + (size_t)(rowBase + lo) * 64;
  v8f acc = {};
#pragma unroll
  for (int kk = 0; kk < 16; ++kk) {
    // A 16x4: VGPR0 = K0(lanes0-15)/K2(lanes16-31); VGPR1 = K1/K3
    const int kb0 = kk * 4 + 2 * hi;
    v2f a, b;
    a.x = xr[kb0];
    a.y = xr[kb0 + 1];
    // B 4x16 mirrors: VGPR0 = K0/K2 rows, N = lane%16
    b.x = W[kb0 * 64 + cc];
    b.y = W[(kb0 + 1) * 64 + cc];
    acc = WMMA_F32_4(a, b, acc);
  }
  const float bv = bias ? bias[cc] : 0.0f;
#pragma unroll
  for (int r = 0; r < 8; ++r)  // C/D: VGPR r -> rows r (lanes0-15), r+8 (lanes16-31)
    Y[(size_t)(rowBase + r + 8 * hi) * 64 + cc] = acc[r] + bv;
}

// ---------------------------------------------------------------------------
// KV path: 8x8/str8 conv over HxW grid -> LayerNorm -> kv projection.
// One block (128 thr) per reduced token; x tile (64 ch x 64 pos) staged in LDS.
// Outputs K,V as [b][h][m][d] contiguous.
// ---------------------------------------------------------------------------
__global__ __launch_bounds__(128) void kv_prep(
    const float* __restrict__ x, const float* __restrict__ sr_w,
    const float* __restrict__ sr_b, const float* __restrict__ ln_g,
    const float* __restrict__ ln_b, const float* __restrict__ kv_w,
    float* __restrict__ kout, float* __restrict__ vout) {
  __shared__ float xt[4096];  // [spatial 0..63][ic 0..63]
  __shared__ float cbuf[64];
  __shared__ float lnv[64];
  __shared__ float stats[2];

  const int tid = threadIdx.x;
  const int b   = blockIdx.x >> 8;
  const int t   = blockIdx.x & 255;
  const int ty  = t >> 4, tx = t & 15;

  for (int i = tid; i < 4096; i += 128) {
    const int s = i >> 6, ic = i & 63;
    const int ky = s >> 3, kx = s & 7;
    const int n = (ty * 8 + ky) * kHW + tx * 8 + kx;
    xt[i] = x[((size_t)b * kN + n) * 64 + ic];
  }
  __syncthreads();

  if (tid < 64) {  // conv: out channel = tid
    const int oc = tid;
    float acc = sr_b[oc];
    const float* wrow = sr_w + (size_t)oc * 4096;  // [oc][ic][ky][kx]
#pragma unroll 4
    for (int ic = 0; ic < 64; ++ic) {
      const float* wi = wrow + ic * 64;  // contiguous over s = ky*8+kx
      const float* xi = xt + ic;
      float a = 0.0f;
      for (int s = 0; s < 64; ++s) a += xi[s * 64] * wi[s];
      acc += a;
    }
    cbuf[oc] = acc;
  }
  __syncthreads();

  if (tid == 0) {  // LN stats over 64 channels (population variance)
    float su = 0.0f, sq = 0.0f;
    for (int i = 0; i < 64; ++i) {
      su += cbuf[i];
      sq += cbuf[i] * cbuf[i];
    }
    const float mu  = su * (1.0f / 64.0f);
    const float var = sq * (1.0f / 64.0f) - mu * mu;
    stats[0] = mu;
    stats[1] = rsqrtf(var + 1e-5f);
  }
  __syncthreads();
  if (tid < 64)
    lnv[tid] = (cbuf[tid] - stats[0]) * stats[1] * ln_g[tid] + ln_b[tid];
  __syncthreads();

  // kv = ln @ kv_w[64,128]; j<64 -> K, j>=64 -> V (dims: [2][heads][hd])
  float acc = 0.0f;
#pragma unroll
  for (int ic = 0; ic < 64; ++ic) acc += lnv[ic] * kv_w[ic * 128 + tid];
  const int j = tid & 63;
  const int h = j >> 3, d = j & 7;
  float* dst = (tid < 64) ? kout : vout;
  dst[(((size_t)b * kHeads + h) * kNK + t) * kHD + d] = acc;
}

// ---------------------------------------------------------------------------
// Fused attention. Block = 64 thr = 2 waves; each wave owns a 16-query strip
// of one (b,h). K/V for the head live in LDS (padded stride 9). Per wave:
//   S[16x256] = scale * Q K^T       : 16 tiles x 2 WMMA (K=4 each)
//   softmax over 256 (2 lanes/row)  : LDS + shfl_xor(16)
//   O[16x8]  = P V                  : 64 chained WMMA (N cols 8..15 zeroed)
// ---------------------------------------------------------------------------
__global__ __launch_bounds__(64) void attn_fused(
    const float* __restrict__ q, const float* __restrict__ kbuf,
    const float* __restrict__ vbuf, float* __restrict__ o) {
  __shared__ float kb[kNK * 9];
  __shared__ float vb[kNK * 9];
  __shared__ float p[2][16 * 260];  // per-wave 16x256 strip, padded stride

  const int tid = threadIdx.x;
  const int bh  = blockIdx.x >> 9;  // 512 query-blocks per (b,h)
  const int qb  = blockIdx.x & 511;
  const int b   = bh >> 3, h = bh & 7;

  const float* Kg = kbuf + (size_t)bh * kNK * kHD;
  const float* Vg = vbuf + (size_t)bh * kNK * kHD;
  for (int i = tid; i < kNK * kHD; i += 64) {
    kb[(i >> 3) * 9 + (i & 7)] = Kg[i];
    vb[(i >> 3) * 9 + (i & 7)] = Vg[i];
  }
  __syncthreads();

  const int wv = tid >> 5, lane = tid & 31, lo = lane & 15, hi = lane >> 4;
  const int q0 = qb * 32 + wv * 16;
  float* pw = p[wv];

  // --- S = scale * Q K^T ---
  const float* qr = q + ((size_t)b * kN + q0 + lo) * 64 + h * kHD;
  v2f a0, a1;  // A operand: row = q0+lo, K pairs (0,1)/(2,3) then (4,5)/(6,7)
  a0.x = qr[2 * hi] * kScale;
  a0.y = qr[2 * hi + 1] * kScale;
  a1.x = qr[4 + 2 * hi] * kScale;
  a1.y = qr[5 + 2 * hi] * kScale;
#pragma unroll
  for (int mt = 0; mt < 16; ++mt) {
    const int m0 = (mt * 16 + lo) * 9;  // B operand: K^T[d][m] = K[m][d]
    v2f b0, b1;
    b0.x = kb[m0 + 2 * hi];
    b0.y = kb[m0 + 2 * hi + 1];
    b1.x = kb[m0 + 4 + 2 * hi];
    b1.y = kb[m0 + 5 + 2 * hi];
    v8f s = {};
    s = WMMA_F32_4(a0, b0, s);
    s = WMMA_F32_4(a1, b1, s);
#pragma unroll
    for (int r = 0; r < 8; ++r)
      pw[(r + 8 * hi) * 260 + mt * 16 + lo] = s[r];
  }

  // --- softmax over 256 columns; lanes (lo, lo+16) split each row ---
  {
    float* pr = pw + lo * 260 + hi * 128;
    float mx = -3.4e38f;
    for (int j = 0; j < 128; ++j) mx = fmaxf(mx, pr[j]);
    mx = fmaxf(mx, __shfl_xor(mx, 16, 32));
    float ss = 0.0f;
    for (int j = 0; j < 128; ++j) {
      const float e = __expf(pr[j] - mx);
      pr[j] = e;
      ss += e;
    }
    ss += __shfl_xor(ss, 16, 32);
    const float inv = 1.0f / ss;
    for (int j = 0; j < 128; ++j) pr[j] *= inv;
  }

  // --- O = P V ---
  v8f ov = {};
#pragma unroll 8
  for (int kk = 0; kk < 64; ++kk) {
    const int m = kk * 4 + 2 * hi;
    v2f pa, vo;
    pa.x = pw[lo * 260 + m];      // A operand: P row lo, K = m..m+1
    pa.y = pw[lo * 260 + m + 1];
    vo.x = (lo < 8) ? vb[m * 9 + lo] : 0.0f;        // B: V[m][d], d = lo
    vo.y = (lo < 8) ? vb[(m + 1) * 9 + lo] : 0.0f;
    ov = WMMA_F32_4(pa, vo, ov);
  }
  if (lo < 8) {
#pragma unroll
    for (int r = 0; r < 8; ++r)
      o[((size_t)b * kN + q0 + r + 8 * hi) * 64 + h * kHD + lo] = ov[r];
  }
}

// ---------------------------------------------------------------------------
extern "C" void kernel_launch(void* const* d_in, const int* in_sizes, int n_in,
                              void* d_out, int out_size, void* d_ws, size_t ws_size,
                              hipStream_t stream) {
  // setup_inputs order: x, H, W, q_w, kv_w, sr_w, sr_b, ln_g, ln_b, proj_w, proj_b
  const float* x      = (const float*)d_in[0];
  const float* q_w    = (const float*)d_in[3];
  const float* kv_w   = (const float*)d_in[4];
  const float* sr_w   = (const float*)d_in[5];
  const float* sr_b   = (const float*)d_in[6];
  const float* ln_g   = (const float*)d_in[7];
  const float* ln_b   = (const float*)d_in[8];
  const float* proj_w = (const float*)d_in[9];
  const float* proj_b = (const float*)d_in[10];
  float* out = (float*)d_out;

  float* ws   = (float*)d_ws;
  float* ws_q = ws;                  // B*N*64 = 2,097,152 floats
  float* ws_o = ws + 2097152;        // B*N*64 = 2,097,152 floats
  float* ws_k = ws + 4194304;        // B*H*Nk*hd = 32,768 floats
  float* ws_v = ws_k + 32768;        // 32,768 floats

  // 1) Q = x @ q_w
  gemm64_wmma<<<(kB * kN) / 16, 128, 0, stream>>>(x, q_w, nullptr, ws_q);
  // 2) conv+LN+kv -> K,V
  kv_prep<<<kB * kNK, 128, 0, stream>>>(x, sr_w, sr_b, ln_g, ln_b, kv_w, ws_k, ws_v);
  // 3) fused attention -> [B,N,C]
  attn_fused<<<kB * kHeads * (kN / 32), 64, 0, stream>>>(ws_q, ws_k, ws_v, ws_o);
  // 4) out = attn_out @ proj_w + proj_b
  gemm64_wmma<<<(kB * kN) / 16, 128, 0, stream>>>(ws_o, proj_w, proj_b, out);
}